// SSD300_init_12652973654620
// MI455X (gfx1250) — compile-verified
//
#include <hip/hip_runtime.h>
#include <hip/hip_bf16.h>
#include <stdint.h>

typedef unsigned u32;
typedef unsigned long long u64;
typedef float v2f __attribute__((ext_vector_type(2)));
typedef float v8f __attribute__((ext_vector_type(8)));

#define BATCH 8
#define NANCH 32768
#define NCLS 81
#define PRE_K 2048
#define MAX_OUT 200
#define IOU_TH 0.5f
#define SCORE_TH 0.05f
#define NWORDS 64              // 2048 bits / 32
#define CH 16                  // scan prefetch chunk rows (16 rows * 256B = 4KB)
#define SORT_CHUNK 2048        // LDS-resident bitonic chunk (16KB of u64 keys)

// ---------------------------------------------------------------------------
// K1: per-anchor class max/argmax -> sortable key + score + category arrays
// key = (~monotone(score) << 32) | anchor_index  => ascending sort gives
// descending score, ties broken by ascending index (matches lax.top_k).
// ---------------------------------------------------------------------------
__global__ __launch_bounds__(256) void k_score(const float* __restrict__ confs,
                                               float* __restrict__ sarr,
                                               int* __restrict__ catarr,
                                               u64* __restrict__ keys) {
  int tid = blockIdx.x * blockDim.x + threadIdx.x;
  if (tid >= BATCH * NANCH) return;
  const float* p = confs + (size_t)tid * NCLS;
  float best = p[0];
  int bc = 0;
  for (int c = 1; c < NCLS; ++c) {
    float v = p[c];
    if (v > best) { best = v; bc = c; }
  }
  bool valid = (best > SCORE_TH) && (bc != 0);
  float s = valid ? best : -1.0f;
  sarr[tid] = s;
  catarr[tid] = bc;
  u32 mb = __float_as_uint(s);
  mb = (mb & 0x80000000u) ? ~mb : (mb | 0x80000000u);  // monotone ascending
  u64 key = ((u64)(~mb) << 32) | (u32)(tid & (NANCH - 1));
  keys[tid] = key;
}

// ---------------------------------------------------------------------------
// K2a: LDS-resident bitonic phase, k = 2 .. 2048, one aligned 2048-key chunk
// per block (16KB LDS; CDNA5 WGP has 320KB). Direction uses the GLOBAL index
// so chunks compose into the cross-chunk merge phases.
// ---------------------------------------------------------------------------
__global__ __launch_bounds__(1024) void k_sort_local(u64* __restrict__ keys) {
  __shared__ u64 sh[SORT_CHUNK];
  int blk = blockIdx.x;
  u64* seg = keys + (size_t)blk * SORT_CHUNK;
  int t = threadIdx.x;
  sh[t] = seg[t];
  sh[t + 1024] = seg[t + 1024];
  __syncthreads();
  int gbase = blk * SORT_CHUNK;
  for (int k = 2; k <= SORT_CHUNK; k <<= 1) {
    for (int j = k >> 1; j > 0; j >>= 1) {
      int i = ((t & ~(j - 1)) << 1) | (t & (j - 1));
      bool up = (((gbase + i) & k) == 0);
      u64 a = sh[i];
      u64 b = sh[i + j];
      if ((a > b) == up) { sh[i] = b; sh[i + j] = a; }
      __syncthreads();
    }
  }
  seg[t] = sh[t];
  seg[t + 1024] = sh[t + 1024];
}

// ---------------------------------------------------------------------------
// K2b: global bitonic exchange (only for j >= 2048; per-batch segments)
// ---------------------------------------------------------------------------
__global__ __launch_bounds__(256) void k_bitonic(u64* __restrict__ keys, int k, int j) {
  int tid = blockIdx.x * blockDim.x + threadIdx.x;
  if (tid >= BATCH * NANCH) return;
  int b = tid >> 15;                // /NANCH
  int i = tid & (NANCH - 1);
  int ixj = i ^ j;
  if (ixj > i) {
    u64* seg = keys + (size_t)b * NANCH;
    u64 a = seg[i];
    u64 c = seg[ixj];
    bool up = ((i & k) == 0);
    if ((a > c) == up) { seg[i] = c; seg[ixj] = a; }
  }
}

// ---------------------------------------------------------------------------
// K2c: LDS-resident merge tail, j = 1024 .. 1 for a given k
// ---------------------------------------------------------------------------
__global__ __launch_bounds__(1024) void k_merge_local(u64* __restrict__ keys, int k) {
  __shared__ u64 sh[SORT_CHUNK];
  int blk = blockIdx.x;
  u64* seg = keys + (size_t)blk * SORT_CHUNK;
  int t = threadIdx.x;
  sh[t] = seg[t];
  sh[t + 1024] = seg[t + 1024];
  __syncthreads();
  int gbase = blk * SORT_CHUNK;
  for (int j = 1024; j > 0; j >>= 1) {
    int i = ((t & ~(j - 1)) << 1) | (t & (j - 1));
    bool up = (((gbase + i) & k) == 0);
    u64 a = sh[i];
    u64 b = sh[i + j];
    if ((a > b) == up) { sh[i] = b; sh[i + j] = a; }
    __syncthreads();
  }
  seg[t] = sh[t];
  seg[t + 1024] = sh[t + 1024];
}

// ---------------------------------------------------------------------------
// K3: gather top-2048 candidates: original boxes, class-offset boxes,
// scores, categories.
// ---------------------------------------------------------------------------
__global__ __launch_bounds__(256) void k_gather(const float* __restrict__ boxes,
                                                const float* __restrict__ sarr,
                                                const int* __restrict__ catarr,
                                                const u64* __restrict__ keys,
                                                float4* __restrict__ tbo,
                                                float4* __restrict__ tbox,
                                                float* __restrict__ tscore,
                                                int* __restrict__ tcat) {
  int tid = blockIdx.x * blockDim.x + threadIdx.x;
  if (tid >= BATCH * PRE_K) return;
  int b = tid >> 11;                // /PRE_K
  int t = tid & (PRE_K - 1);
  u64 key = keys[(size_t)b * NANCH + t];
  int idx = (int)(u32)key;
  const float* bp = boxes + ((size_t)b * NANCH + idx) * 4;
  float4 bx;
  bx.x = bp[0]; bx.y = bp[1]; bx.z = bp[2]; bx.w = bp[3];
  int c = catarr[b * NANCH + idx];
  float s = sarr[b * NANCH + idx];
  float off = 2.0f * (float)c;      // class-aware batched_nms offset
  float4 bo;
  bo.x = bx.x + off; bo.y = bx.y + off; bo.z = bx.z + off; bo.w = bx.w + off;
  tbox[tid] = bx;
  tbo[tid] = bo;
  tscore[tid] = s;
  tcat[tid] = c;
}

// ---------------------------------------------------------------------------
// K4a: zero the suppression bit matrix
// ---------------------------------------------------------------------------
__global__ __launch_bounds__(256) void k_zero(u32* __restrict__ p, int n) {
  int t = blockIdx.x * blockDim.x + threadIdx.x;
  if (t < n) p[t] = 0u;
}

// ---------------------------------------------------------------------------
// K4: suppression bit matrix. One wave per 16x16 tile (lower triangle).
// v_wmma_f32_16x16x4_f32 computes the outer sum area_i + area_j (rank-2
// matmul: A=[area_i,1], B=[1;area_j]); VALU computes the intersection.
// Ballot packs each row's 16 comparison bits; lane 0 ORs into the matrix.
// ---------------------------------------------------------------------------
__global__ __launch_bounds__(256) void k_tiles(const float4* __restrict__ tbo,
                                               u32* __restrict__ supp) {
  int wid = (blockIdx.x * blockDim.x + threadIdx.x) >> 5;  // wave-uniform
  int lane = threadIdx.x & 31;
  int b = wid >> 14;               // 128*128 tiles per batch
  int rem = wid & 16383;
  int ti = rem >> 7;
  int tj = rem & 127;
  if (tj > ti) return;             // wave-uniform exit; EXEC stays all-ones
  int i0 = ti << 4, j0 = tj << 4;
  const float4* base = tbo + (size_t)b * PRE_K;
  int l = lane & 15;

  float4 cb = base[j0 + l];        // this lane's column box (dup on hi half)
  float carea = fmaxf(cb.z - cb.x, 0.0f) * fmaxf(cb.w - cb.y, 0.0f);

  v2f A, Bv;
  if (lane < 16) {                 // A: lanes0-15 hold K=0,1 ; lanes16-31 K=2,3 (zero)
    float4 rb = base[i0 + lane];
    float rarea = fmaxf(rb.z - rb.x, 0.0f) * fmaxf(rb.w - rb.y, 0.0f);
    A.x = rarea; A.y = 1.0f;       // A[i,0]=area_i, A[i,1]=1
    Bv.x = 1.0f; Bv.y = carea;     // B[0,j]=1,      B[1,j]=area_j
  } else {
    A.x = 0.0f; A.y = 0.0f;
    Bv.x = 0.0f; Bv.y = 0.0f;
  }
  v8f C = {};
  // D[i][j] = area_i + area_j
  v8f D = __builtin_amdgcn_wmma_f32_16x16x4_f32(false, A, false, Bv,
                                                (short)0, C, false, false);

  int rbase = i0 + ((lane >> 4) << 3);  // rows 0-7 (lanes<16) or 8-15 (lanes>=16)
  u32 ball[8];
#pragma unroll
  for (int r = 0; r < 8; ++r) {
    float4 rb = base[rbase + r];
    float iw = fmaxf(fminf(rb.z, cb.z) - fmaxf(rb.x, cb.x), 0.0f);
    float ih = fmaxf(fminf(rb.w, cb.w) - fmaxf(rb.y, cb.y), 0.0f);
    float inter = iw * ih;
    float uni = D[r] - inter;
    float iou = inter / fmaxf(uni, 1e-9f);
    ball[r] = (u32)__ballot(iou > IOU_TH);  // lo16: row r, hi16: row r+8
  }
  if (lane == 0) {
    int word = j0 >> 5;
    int shift = j0 & 31;           // 0 or 16
    u32* rowbase = supp + ((size_t)b * PRE_K + i0) * NWORDS + word;
#pragma unroll
    for (int r = 0; r < 8; ++r) {
      atomicOr(rowbase + (size_t)r * NWORDS, (ball[r] & 0xFFFFu) << shift);
      atomicOr(rowbase + (size_t)(r + 8) * NWORDS, (ball[r] >> 16) << shift);
    }
  }
}

// ---------------------------------------------------------------------------
// K5: greedy serial NMS scan. One wave per batch. Keep bitset lives in
// registers (lane l owns words l, l+32), validity bits precomputed into two
// registers per lane so the serial chain has NO global loads. Bit-matrix
// rows are double-buffer prefetched into LDS with
// global_load_async_to_lds_b128 / s_wait_asynccnt.
// ---------------------------------------------------------------------------
__global__ __launch_bounds__(32) void k_scan(const u32* __restrict__ supp,
                                             const float* __restrict__ tscore,
                                             u32* __restrict__ keepmask) {
  int b = blockIdx.x;
  int lane = threadIdx.x;
  __shared__ u32 buf[2][CH * NWORDS];    // 2 x 4KB
  const u32* sbase = supp + (size_t)b * PRE_K * NWORDS;

  auto issue = [&](int c, int slot) {
    u32 ldsbase = (u32)(uintptr_t)(&buf[slot][0]);
    const char* g = (const char*)(sbase + (size_t)c * CH * NWORDS);
#pragma unroll
    for (int t2 = 0; t2 < 8; ++t2) {     // 8 x b128 per lane = 4KB per chunk
      u32 off = (u32)((t2 * 32 + lane) * 16);
      u32 lds = ldsbase + off;
      u64 ga = (u64)(uintptr_t)(g + off);
      asm volatile("global_load_async_to_lds_b128 %0, %1, off"
                   :: "v"(lds), "v"(ga) : "memory");
    }
  };

  issue(0, 0);
  issue(1, 1);

  // Per-lane validity masks for the two owned keep-words (one-time loads).
  int w0 = lane, w1 = lane + 32;
  u32 vm0 = 0, vm1 = 0;
  const float* sc = tscore + (size_t)b * PRE_K;
#pragma unroll 4
  for (int bit = 0; bit < 32; ++bit) {
    vm0 |= (sc[(w0 << 5) + bit] > SCORE_TH ? 1u : 0u) << bit;
    vm1 |= (sc[(w1 << 5) + bit] > SCORE_TH ? 1u : 0u) << bit;
  }

  u32 k0 = 0, k1 = 0;
  const int NCHUNK = PRE_K / CH;         // 128

  for (int c = 0; c < NCHUNK; ++c) {
    if (c + 1 < NCHUNK)
      asm volatile("s_wait_asynccnt 0x8" ::: "memory");  // chunk c resident
    else
      asm volatile("s_wait_asynccnt 0x0" ::: "memory");  // final chunk
    int slot = c & 1;
#pragma unroll 1
    for (int r = 0; r < CH; ++r) {
      int i = c * CH + r;
      int wi = i >> 5, bi = i & 31;
      u32 row0 = buf[slot][r * NWORDS + w0];
      u32 row1 = buf[slot][r * NWORDS + w1];
      u32 a0 = (w0 < wi) ? 0xFFFFFFFFu : ((w0 == wi) ? ((1u << bi) - 1u) : 0u);
      u32 a1 = (w1 < wi) ? 0xFFFFFFFFu : ((w1 == wi) ? ((1u << bi) - 1u) : 0u);
      u32 hit = (row0 & k0 & a0) | (row1 & k1 & a1);
      bool sup = __any((int)(hit != 0u));
      if (wi == w0) {
        if (((vm0 >> bi) & 1u) && !sup) k0 |= 1u << bi;
      } else if (wi == w1) {
        if (((vm1 >> bi) & 1u) && !sup) k1 |= 1u << bi;
      }
    }
    if (c + 2 < NCHUNK) issue(c + 2, slot);
  }
  keepmask[b * NWORDS + w0] = k0;
  keepmask[b * NWORDS + w1] = k1;
}

// ---------------------------------------------------------------------------
// K6: compact first 200 kept entries (already in descending-score order),
// write [B,200,5] boxes+score then [B,200] categories (as float).
// ---------------------------------------------------------------------------
__global__ __launch_bounds__(32) void k_output(const u32* __restrict__ keepmask,
                                               const float4* __restrict__ tbox,
                                               const float* __restrict__ tscore,
                                               const int* __restrict__ tcat,
                                               float* __restrict__ out) {
  int b = blockIdx.x;
  int lane = threadIdx.x;
  float* o5 = out + (size_t)b * MAX_OUT * 5;
  float* oc = out + (size_t)BATCH * MAX_OUT * 5 + (size_t)b * MAX_OUT;
  for (int t = lane; t < MAX_OUT * 5; t += 32) o5[t] = 0.0f;
  for (int t = lane; t < MAX_OUT; t += 32) oc[t] = 0.0f;
  // same-wave store->store to the same address stays ordered
  if (lane == 0) {
    int r = 0;
    for (int w = 0; w < NWORDS && r < MAX_OUT; ++w) {
      u32 m = keepmask[b * NWORDS + w];
      while (m && r < MAX_OUT) {
        int bit = __ffs((int)m) - 1;
        m &= m - 1u;
        int i = (w << 5) + bit;
        float4 bx = tbox[(size_t)b * PRE_K + i];
        o5[r * 5 + 0] = bx.x;
        o5[r * 5 + 1] = bx.y;
        o5[r * 5 + 2] = bx.z;
        o5[r * 5 + 3] = bx.w;
        o5[r * 5 + 4] = tscore[(size_t)b * PRE_K + i];
        oc[r] = (float)tcat[(size_t)b * PRE_K + i];
        ++r;
      }
    }
  }
}

// ---------------------------------------------------------------------------
// Workspace layout (bytes): total ~9.1 MB
// ---------------------------------------------------------------------------
#define WS_KEYS   0                                   // u64[B*N]     2 MB
#define WS_SARR   (WS_KEYS + BATCH * NANCH * 8)       // f32[B*N]     1 MB
#define WS_CAT    (WS_SARR + BATCH * NANCH * 4)       // i32[B*N]     1 MB
#define WS_TBO    (WS_CAT + BATCH * NANCH * 4)        // f4[B*2048] 256 KB
#define WS_TBOX   (WS_TBO + BATCH * PRE_K * 16)       // f4[B*2048] 256 KB
#define WS_TSC    (WS_TBOX + BATCH * PRE_K * 16)      // f32         64 KB
#define WS_TCAT   (WS_TSC + BATCH * PRE_K * 4)        // i32         64 KB
#define WS_SUPP   (WS_TCAT + BATCH * PRE_K * 4)       // u32 bitmat   4 MB
#define WS_KEEP   (WS_SUPP + (size_t)BATCH * PRE_K * NWORDS * 4)

extern "C" void kernel_launch(void* const* d_in, const int* in_sizes, int n_in,
                              void* d_out, int out_size, void* d_ws, size_t ws_size,
                              hipStream_t stream) {
  const float* boxes = (const float*)d_in[0];   // [B,N,4] f32
  const float* confs = (const float*)d_in[1];   // [B,N,81] f32
  float* out = (float*)d_out;                   // [B,200,5] ++ [B,200]

  char* ws = (char*)d_ws;
  u64* keys = (u64*)(ws + WS_KEYS);
  float* sarr = (float*)(ws + WS_SARR);
  int* catarr = (int*)(ws + WS_CAT);
  float4* tbo = (float4*)(ws + WS_TBO);
  float4* tbox = (float4*)(ws + WS_TBOX);
  float* tscore = (float*)(ws + WS_TSC);
  int* tcat = (int*)(ws + WS_TCAT);
  u32* supp = (u32*)(ws + WS_SUPP);
  u32* keepmask = (u32*)(ws + WS_KEEP);

  const int total = BATCH * NANCH;
  const int nchunks = total / SORT_CHUNK;       // 128 LDS sort blocks

  // K1: class reduction + sortable keys
  k_score<<<(total + 255) / 256, 256, 0, stream>>>(confs, sarr, catarr, keys);

  // K2: per-batch bitonic sort (ascending key == descending score).
  // LDS phase covers k=2..2048; only j>=2048 exchanges touch HBM/L2.
  k_sort_local<<<nchunks, 1024, 0, stream>>>(keys);
  for (int k = 2 * SORT_CHUNK; k <= NANCH; k <<= 1) {
    for (int j = k >> 1; j >= SORT_CHUNK; j >>= 1)
      k_bitonic<<<(total + 255) / 256, 256, 0, stream>>>(keys, k, j);
    k_merge_local<<<nchunks, 1024, 0, stream>>>(keys, k);
  }

  // K3: gather top-2048 candidates
  k_gather<<<(BATCH * PRE_K + 255) / 256, 256, 0, stream>>>(
      boxes, sarr, catarr, keys, tbo, tbox, tscore, tcat);

  // K4: suppression bit matrix (zero, then WMMA tile kernel)
  int suppN = BATCH * PRE_K * NWORDS;
  k_zero<<<(suppN + 255) / 256, 256, 0, stream>>>(supp, suppN);
  int nwaves = BATCH * 128 * 128;               // one wave per 16x16 tile
  k_tiles<<<nwaves * 32 / 256, 256, 0, stream>>>(tbo, supp);

  // K5: serial greedy scan, async-LDS double-buffered rows
  k_scan<<<BATCH, 32, 0, stream>>>(supp, tscore, keepmask);

  // K6: compact + write outputs
  k_output<<<BATCH, 32, 0, stream>>>(keepmask, tbox, tscore, tcat, out);
}